// fSRN_56332791054836
// MI455X (gfx1250) — compile-verified
//
#include <hip/hip_runtime.h>
#include <hip/hip_bf16.h>

// ---------------------------------------------------------------------------
// SIREN + forward-mode Jacobian (rotation-free + divergence-free field).
// Compute-bound: ~103 GFLOP fp32 over ~3 MB of unique traffic -> feed the
// matrix pipes. fp32 WMMA (V_WMMA_F32_16X16X4_F32) keeps full precision for
// the sin(30x) chain. Transposed layout (features = M, points/streams = N):
//   every layer is  W(128x128) @ Act^T(128x128)  and the cos-coupling between
//   the h-stream and tangent streams is lane-aligned register math.
// ---------------------------------------------------------------------------

#define OMEGA 30.0f
#define FDIM  128
#define NBLK  3
#define PTS   32            // points per workgroup
#define COLS  128           // 4 streams * 32 points
#define RP    130           // padded row length (floats) for LDS act buffers

typedef __attribute__((ext_vector_type(2))) float v2f;
typedef __attribute__((ext_vector_type(8))) float v8f;

// One GEMM layer:  acc[n-tile] = A(16x128 weights) x B(128x16 activations),
// then elementwise sin/cos coupling. n=0,1 are the h-stream tiles (produce
// sin + keep g = OMEGA*s*cos in regs); n=2..7 are the three tangent streams
// (multiply by the matching g tile). RMW=false writes dst; RMW=true adds the
// residual update into the carry buffer.
template <bool RMW>
__device__ __forceinline__ void layer_gemm(
    const float* __restrict__ W, const float* __restrict__ bvec,
    const float* src, float* dstOrCarry,
    float sscale, int m_base, int lh, int kh)
{
    // A fragments for this wave's 16 feature rows, all K (64 VGPRs).
    // Lane layout for 16x4 f32 A: lanes 0-15 hold K=4kk..4kk+1 of row M=lane,
    // lanes 16-31 hold K=4kk+2..4kk+3 of row M=lane-16 -> one b64 per lane.
    const float* wp = W + (m_base + lh) * FDIM + 2 * kh;
    v2f A[32];
#pragma unroll
    for (int kk = 0; kk < 32; ++kk)
        A[kk] = *(const v2f*)(wp + 4 * kk);

    float bias[8];
#pragma unroll
    for (int v = 0; v < 8; ++v)
        bias[v] = bvec[m_base + v + 8 * kh];

    float g[2][8];  // OMEGA*s*cos(OMEGA*u) for the two h-stream tiles

#pragma unroll
    for (int n = 0; n < 8; ++n) {
        v8f acc = {};
        const float* bp = src + (n * 16 + lh) * RP + 2 * kh;
#pragma unroll
        for (int kk = 0; kk < 32; ++kk) {
            v2f B = *(const v2f*)(bp + 4 * kk);   // ds_load_b64
            acc = __builtin_amdgcn_wmma_f32_16x16x4_f32(
                false, A[kk], false, B, (short)0, acc, false, false);
        }
        const int c = n * 16 + lh;                // column (point/stream)
        if (n < 2) {
            // h-stream: z = s*(h@W^T) + b ; emit sin, keep scaled cos
#pragma unroll
            for (int v = 0; v < 8; ++v) {
                float z  = sscale * acc[v] + bias[v];
                float s1 = __sinf(OMEGA * z);
                g[n][v]  = OMEGA * sscale * __cosf(OMEGA * z);
                int feat = m_base + v + 8 * kh;
                if (RMW) dstOrCarry[c * RP + feat] += s1;
                else     dstOrCarry[c * RP + feat]  = s1;
            }
        } else {
            const int q = n & 1;                  // which 16-point half
#pragma unroll
            for (int v = 0; v < 8; ++v) {
                float d  = g[q][v] * acc[v];
                int feat = m_base + v + 8 * kh;
                if (RMW) dstOrCarry[c * RP + feat] += d;
                else     dstOrCarry[c * RP + feat]  = d;
            }
        }
    }
}

__global__ __launch_bounds__(256)
void siren_curl_kernel(const float* __restrict__ coords,
                       const float* __restrict__ first_w,
                       const float* __restrict__ first_b,
                       const float* __restrict__ res_w1,
                       const float* __restrict__ res_b1,
                       const float* __restrict__ res_w2,
                       const float* __restrict__ res_b2,
                       const float* __restrict__ final_w,
                       float* __restrict__ out)
{
    __shared__ float actC[COLS * RP];   // carry: [col][feat], col = s*32 + p
    __shared__ float actT[COLS * RP];   // intermediate s1/ds1

    const int tid   = threadIdx.x;      // 256 threads = 8 waves (wave32)
    const int lane  = tid & 31;
    const int wave  = tid >> 5;
    const int m_base = wave * 16;
    const int lh = lane & 15;
    const int kh = lane >> 4;
    const int pbase = blockIdx.x * PTS;

    // ---- first layer (K=3): fill h and the three tangents ----
    for (int idx = tid; idx < PTS * FDIM; idx += 256) {
        int p = idx >> 7, j = idx & 127;
        float x0 = coords[(pbase + p) * 3 + 0];
        float x1 = coords[(pbase + p) * 3 + 1];
        float x2 = coords[(pbase + p) * 3 + 2];
        float w0 = first_w[j * 3 + 0];
        float w1 = first_w[j * 3 + 1];
        float w2 = first_w[j * 3 + 2];
        float z  = x0 * w0 + x1 * w1 + x2 * w2 + first_b[j];
        float gc = OMEGA * __cosf(OMEGA * z);
        actC[(0 * PTS + p) * RP + j] = __sinf(OMEGA * z);
        actC[(1 * PTS + p) * RP + j] = gc * w0;
        actC[(2 * PTS + p) * RP + j] = gc * w1;
        actC[(3 * PTS + p) * RP + j] = gc * w2;
    }
    __syncthreads();

    // ---- three residual blocks, 2 GEMM layers each ----
    for (int blk = 0; blk < NBLK; ++blk) {
        const float s = (blk == 0) ? 1.0f : 0.5f;
        // warm L2/L1 for the second matmul's weight rows while GEMM1 runs
        __builtin_prefetch(res_w2 + blk * FDIM * FDIM + (m_base + lh) * FDIM, 0, 0);
        layer_gemm<false>(res_w1 + blk * FDIM * FDIM, res_b1 + blk * FDIM,
                          actC, actT, s, m_base, lh, kh);
        __syncthreads();
        layer_gemm<true>(res_w2 + blk * FDIM * FDIM, res_b2 + blk * FDIM,
                         actT, actC, 1.0f, m_base, lh, kh);
        __syncthreads();
    }

    // ---- final layer: only the Jacobian combos are needed ----
    // out0 = dh0.f0 + dh1.f3 - dh2.f2
    // out1 = dh1.f0 + dh2.f1 - dh0.f3
    // out2 = dh2.f0 + dh0.f2 - dh1.f1
    if (tid < PTS * 3) {
        int p = tid / 3, comp = tid % 3;
        const float* d0 = &actC[(1 * PTS + p) * RP];
        const float* d1 = &actC[(2 * PTS + p) * RP];
        const float* d2 = &actC[(3 * PTS + p) * RP];
        const float* f0 = final_w + 0 * FDIM;
        const float* f1 = final_w + 1 * FDIM;
        const float* f2 = final_w + 2 * FDIM;
        const float* f3 = final_w + 3 * FDIM;
        float acc = 0.0f;
        if (comp == 0) {
            for (int j = 0; j < FDIM; ++j)
                acc += d0[j] * f0[j] + d1[j] * f3[j] - d2[j] * f2[j];
        } else if (comp == 1) {
            for (int j = 0; j < FDIM; ++j)
                acc += d1[j] * f0[j] + d2[j] * f1[j] - d0[j] * f3[j];
        } else {
            for (int j = 0; j < FDIM; ++j)
                acc += d2[j] * f0[j] + d0[j] * f2[j] - d1[j] * f1[j];
        }
        out[(pbase + p) * 3 + comp] = acc;
    }
}

extern "C" void kernel_launch(void* const* d_in, const int* in_sizes, int n_in,
                              void* d_out, int out_size, void* d_ws, size_t ws_size,
                              hipStream_t stream)
{
    const float* coords  = (const float*)d_in[0];
    const float* first_w = (const float*)d_in[1];
    const float* first_b = (const float*)d_in[2];
    const float* res_w1  = (const float*)d_in[3];
    const float* res_b1  = (const float*)d_in[4];
    const float* res_w2  = (const float*)d_in[5];
    const float* res_b2  = (const float*)d_in[6];
    const float* final_w = (const float*)d_in[7];
    float* out = (float*)d_out;

    const int npts = in_sizes[0] / 3;       // 131072
    const int grid = npts / PTS;            // 4096 workgroups of 256 threads
    siren_curl_kernel<<<grid, 256, 0, stream>>>(
        coords, first_w, first_b, res_w1, res_b1, res_w2, res_b2, final_w, out);
}